// DeepLSTM_85581518340764
// MI455X (gfx1250) — compile-verified
//
#include <hip/hip_runtime.h>
#include <hip/hip_bf16.h>

typedef __attribute__((ext_vector_type(2))) float v2f;
typedef __attribute__((ext_vector_type(8))) float v8f;

namespace {
constexpr int kHid   = 512;
constexpr int kBatch = 64;
constexpr int kT     = 1000;
constexpr int kEnt   = 550;
constexpr int kChunk = 100;          // timesteps per precomputed x-gate chunk
constexpr int kG     = 2048;         // 4*kHid gate width
constexpr int kState = kBatch * kHid;   // 32768 floats

// Packed LSTM weight: per 16-wide N tile, K-pairs of 32 floats:
//   Wp[((tile*(K/2) + k/2) * 32) + (n%16)*2 + (k%2)]
// -> one b64 load yields B[k][n], B[k+1][n] for lane (hi selects k+2*hi).
constexpr size_t SZ_WP    = 128ull * 512 * 32;          // 1024x2048 -> 2,097,152
constexpr size_t SZ_WOUTP = 35ull * 256 * 32;           // 512x560   ->   286,720
constexpr size_t SZ_XG0   = (size_t)kBatch * kChunk * kG;

constexpr size_t OFF_W0P   = 0;
constexpr size_t OFF_W1P   = OFF_W0P + SZ_WP;
constexpr size_t OFF_WOUTP = OFF_W1P + SZ_WP;
constexpr size_t OFF_XG0   = OFF_WOUTP + SZ_WOUTP;
constexpr size_t OFF_Y0    = OFF_XG0 + SZ_XG0;          // ping-pong [2][64*512]
constexpr size_t OFF_H0    = OFF_Y0 + 2 * (size_t)kState;
constexpr size_t OFF_C0    = OFF_H0 + 2 * (size_t)kState;
constexpr size_t OFF_H1    = OFF_C0 + 2 * (size_t)kState;
constexpr size_t OFF_C1    = OFF_H1 + 2 * (size_t)kState;
constexpr size_t OFF_LAST  = OFF_C1 + 2 * (size_t)kState;
constexpr size_t ZERO_N    = 11 * (size_t)kState;       // y0,h0,c0,h1,c1 (x2) + last
}  // namespace

static __device__ __forceinline__ v8f wmma4(v2f a, v2f b, v8f c) {
  // V_WMMA_F32_16X16X4_F32: exact f32, D = A(16x4) * B(4x16) + C
  return __builtin_amdgcn_wmma_f32_16x16x4_f32(false, a, false, b, (short)0, c,
                                               false, false);
}

static __device__ __forceinline__ float sigm(float x) {
  return 1.0f / (1.0f + __expf(-x));
}

// ---------------------------------------------------------------- utilities
__global__ void zero_kernel(float* __restrict__ p, size_t n) {
  size_t i = (size_t)blockIdx.x * blockDim.x + threadIdx.x;
  if (i < n) p[i] = 0.0f;
}

__global__ void pack_lstm_w(const float* __restrict__ W, float* __restrict__ Wp) {
  int idx = blockIdx.x * blockDim.x + threadIdx.x;  // over 1024*2048
  if (idx >= 1024 * kG) return;
  int k = idx / kG, n = idx - k * kG;
  int tile = n >> 4, nn = n & 15;
  size_t dst = ((size_t)tile * 512 + (k >> 1)) * 32 + (nn << 1) + (k & 1);
  Wp[dst] = W[idx];
}

__global__ void pack_wout(const float* __restrict__ W, float* __restrict__ Wp) {
  int idx = blockIdx.x * blockDim.x + threadIdx.x;  // over 512*560 (N padded)
  if (idx >= 512 * 560) return;
  int k = idx / 560, n = idx - k * 560;
  int tile = n >> 4, nn = n & 15;
  float v = (n < kEnt) ? W[(size_t)k * kEnt + n] : 0.0f;
  size_t dst = ((size_t)tile * 256 + (k >> 1)) * 32 + (nn << 1) + (k & 1);
  Wp[dst] = v;
}

// ------------------------------------------------ layer-0 x-part (parallel)
// XG0[b*kChunk+t][n] = emb[tok[b, base+t]] @ W0x + b0   (400 M-tiles x 128 N-tiles)
__global__ void __launch_bounds__(256) xg0_kernel(
    const int* __restrict__ inputs, const float* __restrict__ emb,
    const float* __restrict__ W0p, const float* __restrict__ b0,
    float* __restrict__ XG0, int chunk_base) {
  int wave = threadIdx.x >> 5, lane = threadIdx.x & 31;
  int gw = blockIdx.x * 8 + wave;            // 0..51199
  int tile_m = gw >> 7, tile_n = gw & 127;
  int hi = lane >> 4, nn = lane & 15;

  int mA = (tile_m << 4) + nn;               // A-row for this lane (M = lane%16)
  int b = mA / kChunk, tl = mA - b * kChunk;
  int tok = inputs[b * kT + chunk_base + tl];
  const float* arow = emb + (size_t)tok * kHid;

  int n0 = tile_n << 4;
  float bias = b0[n0 + nn];
  v8f acc;
#pragma unroll
  for (int r = 0; r < 8; ++r) acc[r] = bias;

  const float* bt = W0p + (size_t)tile_n * (512 * 32);  // x-part: kpairs 0..255
  for (int kc = 0; kc < kHid; kc += 4) {
    v2f a = *(const v2f*)(arow + kc + 2 * hi);
    v2f bv = *(const v2f*)(bt + ((size_t)((kc >> 1) + hi) << 5) + (nn << 1));
    acc = wmma4(a, bv, acc);
  }
  float* orow = XG0 + (size_t)(tile_m << 4) * kG + n0 + nn;
#pragma unroll
  for (int r = 0; r < 8; ++r) orow[(size_t)(r + 8 * hi) * kG] = acc[r];
}

// --------------------------------------------------------- LSTM cell update
static __device__ __forceinline__ void cell_update(
    const v8f* acc /*i,j,f,o*/, int m0, int hi, int nn, int u0,
    const int* __restrict__ lengths, int t,
    const float* __restrict__ crd, const float* __restrict__ hrd,
    float* __restrict__ cwr, float* __restrict__ hwr,
    float* __restrict__ ywr, float* __restrict__ lastp) {
  int u = u0 + nn;
#pragma unroll
  for (int r = 0; r < 8; ++r) {
    int mr = m0 + r + 8 * hi;                 // batch row for this C component
    int len = lengths[mr];
    size_t idx = (size_t)mr * kHid + u;
    float cold = crd[idx], hold = hrd[idx];
    float ii = sigm(acc[0][r]);
    float jj = tanhf(acc[1][r]);
    float ff = sigm(acc[2][r]);
    float oo = sigm(acc[3][r]);
    float cn = ff * cold + ii * jj;
    float hn = oo * tanhf(cn);
    bool act = t < len;
    cwr[idx] = act ? cn : cold;
    hwr[idx] = act ? hn : hold;
    if (ywr) ywr[idx] = act ? hn : 0.0f;
    if (lastp && (t + 1 == len)) lastp[idx] = hn;  // h2 at t = len-1
  }
}

// ------------------------------------------- pipelined recurrent step kernel
// super-step s: layer0 computes t=s (if s<T); layer1 computes t=s-1 (if s>=1).
// Split-K x4: a group of 4 waves shares one 16(batch) x 16(unit) tile; each
// wave covers a quarter of K (all 4 gates), partials reduced through LDS,
// split-0 wave performs the fused cell update. 2 tiles per 256-thread block.
__global__ void __launch_bounds__(256) step_kernel(
    int s, const float* __restrict__ XG0, const float* __restrict__ W0p,
    const float* __restrict__ W1p, const float* __restrict__ b1,
    const int* __restrict__ lengths, float* __restrict__ h0b,
    float* __restrict__ c0b, float* __restrict__ h1b, float* __restrict__ c1b,
    float* __restrict__ y0b, float* __restrict__ lastp) {
  __shared__ v8f red[2][3][4][32];           // [tile][split-1][gate][lane]
  int wave = threadIdx.x >> 5, lane = threadIdx.x & 31;
  int tb = wave >> 2, ks = wave & 3;         // tile-in-block, K-split id
  int tile = blockIdx.x * 2 + tb;            // 0..255
  int layer = tile >> 7;                     // blocks are single-layer
  int tid = tile & 127;                      // 4 M-tiles x 32 U-tiles
  int tile_m = tid >> 5, tile_u = tid & 31;
  int hi = lane >> 4, nn = lane & 15;
  int m0 = tile_m << 4, u0 = tile_u << 4;
  int mA = m0 + nn;

  if (layer == 0 && s >= kT) return;         // block-uniform exits
  if (layer == 1 && s < 1) return;

  const float *crd, *hrd, *abase, *Wp;
  float *cwr, *hwr, *ywr, *lastw;
  int t, kspan, kpair0;
  v8f acc[4];

  if (layer == 0) {
    t = s;
    hrd = h0b + (size_t)(t & 1) * kState;
    hwr = h0b + (size_t)((t + 1) & 1) * kState;
    crd = c0b + (size_t)(t & 1) * kState;
    cwr = c0b + (size_t)((t + 1) & 1) * kState;
    ywr = y0b + (size_t)(t & 1) * kState;
    lastw = nullptr;
    Wp = W0p;
    kspan = 128;                             // 512 recurrent K / 4 splits
    kpair0 = 256 + 64 * ks;                  // recurrent kpairs start at 256
    abase = hrd + ks * 128;
    if (ks == 0) {                           // partial 0 seeds with x-gates (+b0)
      int tl = t % kChunk;
#pragma unroll
      for (int g = 0; g < 4; ++g)
#pragma unroll
        for (int r = 0; r < 8; ++r) {
          int mr = m0 + r + 8 * hi;
          acc[g][r] = XG0[(size_t)(mr * kChunk + tl) * kG + g * kHid + u0 + nn];
        }
    } else {
#pragma unroll
      for (int g = 0; g < 4; ++g)
#pragma unroll
        for (int r = 0; r < 8; ++r) acc[g][r] = 0.0f;
    }
  } else {
    t = s - 1;
    const float* xin = y0b + (size_t)(t & 1) * kState;  // layer-0 masked output
    hrd = h1b + (size_t)(t & 1) * kState;
    hwr = h1b + (size_t)((t + 1) & 1) * kState;
    crd = c1b + (size_t)(t & 1) * kState;
    cwr = c1b + (size_t)((t + 1) & 1) * kState;
    ywr = nullptr;
    lastw = lastp;
    Wp = W1p;
    kspan = 256;                             // 1024 concat K / 4 splits
    kpair0 = 128 * ks;                       // x-part kpairs 0..255, h-part 256..511
    abase = ((ks < 2) ? xin : hrd) + (ks & 1) * 256;
    if (ks == 0) {
#pragma unroll
      for (int g = 0; g < 4; ++g) {
        float bias = b1[g * kHid + u0 + nn];
#pragma unroll
        for (int r = 0; r < 8; ++r) acc[g][r] = bias;
      }
    } else {
#pragma unroll
      for (int g = 0; g < 4; ++g)
#pragma unroll
        for (int r = 0; r < 8; ++r) acc[g][r] = 0.0f;
    }
  }

  const float* arow = abase + (size_t)mA * kHid;
  const float* bt0 = Wp + (size_t)tile_u * (512 * 32);
  for (int kc = 0; kc < kspan; kc += 4) {
    v2f a = *(const v2f*)(arow + kc + 2 * hi);
    size_t koff = ((size_t)(kpair0 + (kc >> 1) + hi) << 5) + (nn << 1);
#pragma unroll
    for (int g = 0; g < 4; ++g) {            // gate tile = g*32 + tile_u
      v2f bv = *(const v2f*)(bt0 + (size_t)g * (32 * 512 * 32) + koff);
      acc[g] = wmma4(a, bv, acc[g]);
    }
  }

  if (ks != 0) {                             // publish partials to LDS
#pragma unroll
    for (int g = 0; g < 4; ++g) red[tb][ks - 1][g][lane] = acc[g];
  }
  __syncthreads();
  if (ks == 0) {                             // reduce + fused cell update
#pragma unroll
    for (int w = 0; w < 3; ++w)
#pragma unroll
      for (int g = 0; g < 4; ++g) acc[g] = acc[g] + red[tb][w][g][lane];
    cell_update(acc, m0, hi, nn, u0, lengths, t, crd, hrd, cwr, hwr, ywr,
                lastw);
  }
}

// ----------------------------------------------------------- final projection
__global__ void __launch_bounds__(256) logits_kernel(
    const float* __restrict__ last, const float* __restrict__ Wp,
    const float* __restrict__ bout, float* __restrict__ out) {
  int wave = threadIdx.x >> 5, lane = threadIdx.x & 31;
  int gw = blockIdx.x * 8 + wave;
  if (gw >= 4 * 35) return;
  int tile_m = gw / 35, tile_n = gw - tile_m * 35;
  int hi = lane >> 4, nn = lane & 15;
  int m0 = tile_m << 4, n0 = tile_n << 4;
  int n = n0 + nn;
  float bias = (n < kEnt) ? bout[n] : 0.0f;
  v8f acc;
#pragma unroll
  for (int r = 0; r < 8; ++r) acc[r] = bias;

  const float* arow = last + (size_t)(m0 + nn) * kHid;
  const float* bt = Wp + (size_t)tile_n * (256 * 32);
  for (int kc = 0; kc < kHid; kc += 4) {
    v2f a = *(const v2f*)(arow + kc + 2 * hi);
    v2f bv = *(const v2f*)(bt + ((size_t)((kc >> 1) + hi) << 5) + (nn << 1));
    acc = wmma4(a, bv, acc);
  }
  if (n < kEnt) {
#pragma unroll
    for (int r = 0; r < 8; ++r)
      out[(size_t)(m0 + r + 8 * hi) * kEnt + n] = acc[r];
  }
}

// ---------------------------------------------------------------------------
extern "C" void kernel_launch(void* const* d_in, const int* in_sizes, int n_in,
                              void* d_out, int out_size, void* d_ws,
                              size_t ws_size, hipStream_t stream) {
  (void)in_sizes; (void)n_in; (void)out_size; (void)ws_size;
  const int*   inputs  = (const int*)d_in[0];
  const int*   lengths = (const int*)d_in[1];
  const float* emb     = (const float*)d_in[2];
  const float* W0      = (const float*)d_in[3];
  const float* b0      = (const float*)d_in[4];
  const float* W1      = (const float*)d_in[5];
  const float* b1      = (const float*)d_in[6];
  const float* Wout    = (const float*)d_in[7];
  const float* bout    = (const float*)d_in[8];
  float* out = (float*)d_out;
  float* ws  = (float*)d_ws;

  float* W0p   = ws + OFF_W0P;
  float* W1p   = ws + OFF_W1P;
  float* Woutp = ws + OFF_WOUTP;
  float* XG0   = ws + OFF_XG0;
  float* y0b   = ws + OFF_Y0;
  float* h0b   = ws + OFF_H0;
  float* c0b   = ws + OFF_C0;
  float* h1b   = ws + OFF_H1;
  float* c1b   = ws + OFF_C1;
  float* lastp = ws + OFF_LAST;

  // zero LSTM states / ping-pong buffers / last-h (deterministic per call)
  zero_kernel<<<(unsigned)((ZERO_N + 255) / 256), 256, 0, stream>>>(y0b, ZERO_N);
  // pack weights into WMMA-native tile layout (once per call; L2-resident after)
  pack_lstm_w<<<(1024 * kG + 255) / 256, 256, 0, stream>>>(W0, W0p);
  pack_lstm_w<<<(1024 * kG + 255) / 256, 256, 0, stream>>>(W1, W1p);
  pack_wout<<<(512 * 560 + 255) / 256, 256, 0, stream>>>(Wout, Woutp);

  // pipelined scan: layer0 at t=s, layer1 at t=s-1 (split-K x4 per tile)
  for (int s = 0; s <= kT; ++s) {
    if (s < kT && (s % kChunk) == 0)
      xg0_kernel<<<6400, 256, 0, stream>>>(inputs, emb, W0p, b0, XG0, s);
    step_kernel<<<128, 256, 0, stream>>>(s, XG0, W0p, W1p, b1, lengths, h0b,
                                         c0b, h1b, c1b, y0b, lastp);
  }

  logits_kernel<<<18, 256, 0, stream>>>(lastp, Woutp, bout, out);
}